// MorphologicalTranformation_36876589203631
// MI455X (gfx1250) — compile-verified
//
#include <hip/hip_runtime.h>
#include <cstdint>

// Morphological opening: 11x11 erode (min) then 11x11 dilate (max), per channel.
// (16, 2048, 2048) fp32. Fused separable LDS-tiled kernel.
// CDNA5 paths: TDM tensor_load_to_lds (interior tiles), async global->LDS B128
// (border tiles), all-B128 LDS phases, B128 global stores.

#define C_   16
#define H_   2048
#define W_   2048
#define R_   5                  // radius of the 11x11 structuring element
#define TS   64                 // output tile (TS x TS) per block
#define TC   (TS + 24)          // 88: staged tile cols (12-col halo each side, 16B aligned)
#define TR   (TS + 20)          // 84: staged tile rows (10-row halo each side)
#define CHUNKS_X (TC / 4)       // 22 float4 chunks per staged row
#define EW   76                 // padded h-min/eroded width (74 valid cols, 16B pitch)
#define NTHREADS 256

typedef unsigned int v4u __attribute__((ext_vector_type(4)));
typedef int          v4i __attribute__((ext_vector_type(4)));
typedef int          v8i __attribute__((ext_vector_type(8)));

// CDNA5 async DMA: 16 bytes per active lane, global -> LDS (ASYNCcnt).
__device__ __forceinline__ void async_ld_b128(uint32_t lds_byte_off, const float* g) {
  asm volatile("global_load_async_to_lds_b128 %0, %1, off"
               :: "v"(lds_byte_off), "v"(g)
               : "memory");
}

// CDNA5 Tensor Data Mover: DMA an 88x84 fp32 tile (row-major, packed) from the
// 2048x2048 image into LDS. D# packing per CDNA5 ISA.
__device__ __forceinline__ void tdm_load_tile(const float* gsrc, uint32_t lds_off) {
  const uint64_t ga = (uint64_t)(uintptr_t)gsrc;
  v4u g0;
  g0[0] = 1u;                                            // count=1, user descriptor
  g0[1] = lds_off;                                       // lds_addr (bytes)
  g0[2] = (uint32_t)ga;                                  // global_addr[31:0]
  g0[3] = (uint32_t)((ga >> 32) & 0x1FFFFFFu) | (2u << 30); // global_addr[56:32] | type=2
  v8i g1;
  g1[0] = (int)(2u << 16);                               // data_size = 4 bytes
  g1[1] = (int)((uint32_t)(W_ & 0xFFFF) << 16);          // tensor_dim0[15:0] @ bits 63:48
  g1[2] = (int)((((uint32_t)W_) >> 16) |                 // tensor_dim0[31:16]
                ((uint32_t)(H_ & 0xFFFF) << 16));        // tensor_dim1[15:0]
  g1[3] = (int)((((uint32_t)H_) >> 16) |                 // tensor_dim1[31:16]
                ((uint32_t)TC << 16));                   // tile_dim0 = 88
  g1[4] = (int)(uint32_t)TR;                             // tile_dim1 = 84, tile_dim2 = 0
  g1[5] = (int)(uint32_t)W_;                             // tensor_dim0_stride[31:0] = 2048
  g1[6] = 0;                                             // stride hi / dim1_stride lo
  g1[7] = 0;
  v4i gz = {0, 0, 0, 0};
#if __clang_major__ >= 23
  v8i z8 = {0, 0, 0, 0, 0, 0, 0, 0};
  __builtin_amdgcn_tensor_load_to_lds(g0, g1, gz, gz, z8, 0);
#else
  __builtin_amdgcn_tensor_load_to_lds(g0, g1, gz, gz, 0);
#endif
}

__global__ __launch_bounds__(NTHREADS)
void morph_open_fused(const float* __restrict__ in, float* __restrict__ out) {
  __shared__ __align__(16) float A[TR * TC];  // input tile -> later eroded tile
  __shared__ __align__(16) float B[TR * EW];  // h-min tile -> later h-max tile

  const int tid = threadIdx.x;
  const int ox  = blockIdx.x * TS;
  const int oy  = blockIdx.y * TS;
  const int c   = blockIdx.z;
  const float* img = in + (size_t)c * H_ * W_;

  // ---- Phase 1: stage 84-row x 88-col input tile into LDS.                  ----
  // Rows gy in [oy-10, oy+74), cols gx in [ox-12, ox+76).
  const bool interior = (ox >= 12) && (ox + TS + 12 <= W_) &&
                        (oy >= 10) && (oy + TS + 10 <= H_);
  if (interior) {
    // Whole tile in-image: one TDM descriptor DMAs it (wave 0 issues; TDM
    // ignores EXEC; TENSORcnt wait + barrier publishes the LDS writes).
    if (tid == 0) {
      tdm_load_tile(img + (size_t)(oy - 10) * W_ + (ox - 12), (uint32_t)(uintptr_t)&A[0]);
      __builtin_amdgcn_s_wait_tensorcnt(0);
    }
  } else {
    // Border tile: per-lane async B128 chunks; OOB chunks -> +inf (erode pad).
    // Aligned 4-col chunks are entirely in- or out-of-image (W % 4 == 0).
    for (int i = tid; i < TR * CHUNKS_X; i += NTHREADS) {
      const int ly = i / CHUNKS_X, k = i - ly * CHUNKS_X;
      const int gy = oy + ly - 10;
      const int gx = ox + 4 * k - 12;
      const int lidx = ly * TC + 4 * k;
      if ((unsigned)gy < (unsigned)H_ && (unsigned)gx < (unsigned)W_) {
        async_ld_b128((uint32_t)(uintptr_t)&A[lidx], img + (size_t)gy * W_ + gx);
      } else {
        const float inf = __builtin_inff();
        *(float4*)&A[lidx] = make_float4(inf, inf, inf, inf);
      }
    }
    __builtin_amdgcn_s_wait_asynccnt(0);
  }
  __syncthreads();

  // ---- Phase 2: horizontal min (erode-x): A[84x88] -> B[84 x 76(74 valid)]. ----
  // Output col lx+j maps to window A[ly][lx+j+2 .. lx+j+12]; 4 outputs/thread
  // from a 16-float register window with a shared-min tree.
  for (int i = tid; i < TR * (EW / 4); i += NTHREADS) {
    const int ly = i / (EW / 4), k = i - ly * (EW / 4);
    const float4* rowA = (const float4*)&A[ly * TC];     // pitch 352B, 16B aligned
    const float4 q0 = rowA[k], q1 = rowA[k + 1], q2 = rowA[k + 2], q3 = rowA[k + 3];
    const float w0 = q0.x, w1 = q0.y, w2 = q0.z, w3 = q0.w;
    const float w4 = q1.x, w5 = q1.y, w6 = q1.z, w7 = q1.w;
    const float w8 = q2.x, w9 = q2.y, w10 = q2.z, w11 = q2.w;
    const float w12 = q3.x, w13 = q3.y, w14 = q3.z, w15 = q3.w;
    // common core: min over w5..w12 (shared by all four 11-wide windows)
    float cm = fminf(fminf(fminf(w5, w6), fminf(w7, w8)),
                     fminf(fminf(w9, w10), fminf(w11, w12)));
    float4 o;
    o.x = fminf(fminf(w2, w3),  fminf(w4,  cm));   // window 2..12
    o.y = fminf(fminf(w3, w4),  fminf(w13, cm));   // window 3..13
    o.z = fminf(fminf(w4, w13), fminf(w14, cm));   // window 4..14
    o.w = fminf(fminf(w13, w14), fminf(w15, cm));  // window 5..15
    ((float4*)&B[ly * EW])[k] = o;                 // pitch 304B, 16B aligned
  }
  __syncthreads();

  // ---- Phase 3: vertical min (erode-y) + border mask: B -> A[74 x 76].      ----
  for (int i = tid; i < 74 * (EW / 4); i += NTHREADS) {
    const int ly = i / (EW / 4), k = i - ly * (EW / 4);
    const int lx = 4 * k;
    float4 m = ((const float4*)&B[ly * EW])[k];
#pragma unroll
    for (int d = 1; d <= 2 * R_; ++d) {
      const float4 v = ((const float4*)&B[(ly + d) * EW])[k];
      m.x = fminf(m.x, v.x); m.y = fminf(m.y, v.y);
      m.z = fminf(m.z, v.z); m.w = fminf(m.w, v.w);
    }
    // Eroded values at out-of-image positions must be -inf so they never win
    // the dilation max (exact reduce_window border semantics).
    const int ey = oy + ly - R_;
    const float ninf = -__builtin_inff();
    if ((unsigned)ey >= (unsigned)H_) {
      m = make_float4(ninf, ninf, ninf, ninf);
    } else {
      const int ex = ox + lx - R_;
      if ((unsigned)(ex + 0) >= (unsigned)W_) m.x = ninf;
      if ((unsigned)(ex + 1) >= (unsigned)W_) m.y = ninf;
      if ((unsigned)(ex + 2) >= (unsigned)W_) m.z = ninf;
      if ((unsigned)(ex + 3) >= (unsigned)W_) m.w = ninf;
    }
    ((float4*)&A[ly * EW])[k] = m;
  }
  __syncthreads();

  // ---- Phase 4: horizontal max (dilate-x): A[74x76] -> B[74 x 64].          ----
  // Output col lx+j maps to eroded window [lx+j .. lx+j+10].
  for (int i = tid; i < 74 * (TS / 4); i += NTHREADS) {
    const int ly = i / (TS / 4), k = i - ly * (TS / 4);
    const float4* rowE = (const float4*)&A[ly * EW];
    const float4 q0 = rowE[k], q1 = rowE[k + 1], q2 = rowE[k + 2], q3 = rowE[k + 3];
    const float w0 = q0.x, w1 = q0.y, w2 = q0.z, w3 = q0.w;
    const float w4 = q1.x, w5 = q1.y, w6 = q1.z, w7 = q1.w;
    const float w8 = q2.x, w9 = q2.y, w10 = q2.z, w11 = q2.w;
    const float w12 = q3.x, w13 = q3.y;
    // common core: max over w3..w10
    float cm = fmaxf(fmaxf(fmaxf(w3, w4), fmaxf(w5, w6)),
                     fmaxf(fmaxf(w7, w8), fmaxf(w9, w10)));
    float4 o;
    o.x = fmaxf(fmaxf(w0, w1),  fmaxf(w2,  cm));   // window 0..10
    o.y = fmaxf(fmaxf(w1, w2),  fmaxf(w11, cm));   // window 1..11
    o.z = fmaxf(fmaxf(w2, w11), fmaxf(w12, cm));   // window 2..12
    o.w = fmaxf(fmaxf(w11, w12), fmaxf(w13, cm));  // window 3..13
    ((float4*)&B[ly * TS])[k] = o;                 // pitch 256B
  }
  __syncthreads();

  // ---- Phase 5: vertical max (dilate-y): B -> global output (B128 stores).  ----
  float* outc = out + (size_t)c * H_ * W_;
  for (int i = tid; i < TS * (TS / 4); i += NTHREADS) {
    const int ly = i / (TS / 4), k = i - ly * (TS / 4);
    float4 m = ((const float4*)&B[ly * TS])[k];
#pragma unroll
    for (int d = 1; d <= 2 * R_; ++d) {
      const float4 v = ((const float4*)&B[(ly + d) * TS])[k];
      m.x = fmaxf(m.x, v.x); m.y = fmaxf(m.y, v.y);
      m.z = fmaxf(m.z, v.z); m.w = fmaxf(m.w, v.w);
    }
    *(float4*)&outc[(size_t)(oy + ly) * W_ + (ox + 4 * k)] = m;
  }
}

extern "C" void kernel_launch(void* const* d_in, const int* in_sizes, int n_in,
                              void* d_out, int out_size, void* d_ws, size_t ws_size,
                              hipStream_t stream) {
  (void)in_sizes; (void)n_in; (void)out_size; (void)d_ws; (void)ws_size;
  const float* in  = (const float*)d_in[0];
  float*       out = (float*)d_out;
  dim3 grid(W_ / TS, H_ / TS, C_);   // (32, 32, 16)
  morph_open_fused<<<grid, NTHREADS, 0, stream>>>(in, out);
}